// VectorQuantizer_29145648070707
// MI455X (gfx1250) — compile-verified
//
#include <hip/hip_runtime.h>
#include <hip/hip_bf16.h>

typedef __attribute__((ext_vector_type(2))) float v2f;
typedef __attribute__((ext_vector_type(8))) float v8f;

#define EMBED_DIM 64
#define TOKEN_TILES 2                        // 16-token tiles per wave
#define TOKENS_PER_WAVE (16 * TOKEN_TILES)   // 32
#define WAVES_PER_BLOCK 4
#define TOKENS_PER_BLOCK (TOKENS_PER_WAVE * WAVES_PER_BLOCK)  // 128

// ---------------------------------------------------------------------------
// Kernel 1: per-code squared norms of the codebook rows -> ws
// ---------------------------------------------------------------------------
__global__ void vq_enorm_kernel(const float* __restrict__ codebook,
                                float* __restrict__ enorm, int K) {
    int i = blockIdx.x * blockDim.x + threadIdx.x;
    if (i < K) {
        const float* row = codebook + i * EMBED_DIM;
        float s = 0.f;
#pragma unroll
        for (int d = 0; d < EMBED_DIM; ++d) {
            float v = row[d];
            s += v * v;
        }
        enorm[i] = s;
    }
}

// ---------------------------------------------------------------------------
// Kernel 2: main VQ kernel. One wave32 handles 32 tokens (2 WMMA token tiles,
// register-blocked so each B fragment feeds 2 WMMAs).
//   S = X_tile (16x64) * E^T (64 x 16-code tile) via V_WMMA_F32_16X16X4_F32,
//   dist = ||e||^2 - 2*dot, per-lane running argmin, cross-lane reduce.
// ---------------------------------------------------------------------------
__global__ void __launch_bounds__(128)
vq_main_kernel(const float* __restrict__ x,
               const float* __restrict__ codebook,
               const float* __restrict__ enorm,
               float* __restrict__ out_q,
               float* __restrict__ out_idx,
               float* __restrict__ partials,
               int K) {
    const int lane   = threadIdx.x & 31;
    const int waveId = threadIdx.x >> 5;
    const int tok0   = (blockIdx.x * WAVES_PER_BLOCK + waveId) * TOKENS_PER_WAVE;

    // A-fragment addressing for V_WMMA_F32_16X16X4_F32 (16x4 f32 A layout):
    //   lane L: M = L % 16 ; VGPR v holds K = (L/16)*2 + v
    const int m   = lane & 15;   // token row inside tile
    const int klo = (lane >> 4) * 2;

    // Preload all K-chunks of A for both token tiles (x stays in registers).
    v2f a[TOKEN_TILES][EMBED_DIM / 4];
#pragma unroll
    for (int tt = 0; tt < TOKEN_TILES; ++tt) {
        const float* xrow = x + (tok0 + tt * 16 + m) * EMBED_DIM + klo;
#pragma unroll
        for (int s = 0; s < EMBED_DIM / 4; ++s) {
            a[tt][s] = *(const v2f*)(xrow + s * 4);
        }
    }

    // Running per-lane argmin state: slot j covers token (j + 8*(lane>>4)) of
    // tile tt; this lane sees codes with (code % 16) == (lane & 15).
    float bd[TOKEN_TILES][8];
    int   bi[TOKEN_TILES][8];
#pragma unroll
    for (int tt = 0; tt < TOKEN_TILES; ++tt)
#pragma unroll
        for (int j = 0; j < 8; ++j) { bd[tt][j] = 3.4e38f; bi[tt][j] = 0; }

    const int myCol = lane & 15;
    for (int c0 = 0; c0 < K; c0 += 16) {
        v8f acc[TOKEN_TILES];
#pragma unroll
        for (int tt = 0; tt < TOKEN_TILES; ++tt) acc[tt] = (v8f){};

        // B fragment: B[k][n] = codebook[c0+n][s*4 + k]; same (lane -> k) map as A.
        const float* bbase = codebook + (c0 + myCol) * EMBED_DIM + klo;
#pragma unroll
        for (int s = 0; s < EMBED_DIM / 4; ++s) {
            v2f b = *(const v2f*)(bbase + s * 4);
#pragma unroll
            for (int tt = 0; tt < TOKEN_TILES; ++tt) {
                acc[tt] = __builtin_amdgcn_wmma_f32_16x16x4_f32(
                    false, a[tt][s], false, b, (short)0, acc[tt], false, false);
            }
        }
        const float en = enorm[c0 + myCol];
        const int   c  = c0 + myCol;
#pragma unroll
        for (int tt = 0; tt < TOKEN_TILES; ++tt)
#pragma unroll
            for (int j = 0; j < 8; ++j) {
                float d = en - 2.0f * acc[tt][j];
                if (d < bd[tt][j]) { bd[tt][j] = d; bi[tt][j] = c; }  // ascending c0 => first-min
            }
    }

    // Reduce across the 16 lanes of each half (xor offsets stay within halves).
#pragma unroll
    for (int off = 1; off < 16; off <<= 1) {
#pragma unroll
        for (int tt = 0; tt < TOKEN_TILES; ++tt)
#pragma unroll
            for (int j = 0; j < 8; ++j) {
                float od = __shfl_xor(bd[tt][j], off, 32);
                int   oi = __shfl_xor(bi[tt][j], off, 32);
                if (od < bd[tt][j] || (od == bd[tt][j] && oi < bi[tt][j])) {
                    bd[tt][j] = od; bi[tt][j] = oi;
                }
            }
    }
    // Lanes 0-15 agree on bests for tokens 0..7 of each tile (slot j),
    // lanes 16-31 on tokens 8..15.

    // Emit outputs + accumulate squared error for the loss.
    float lsum = 0.f;
#pragma unroll
    for (int tt = 0; tt < TOKEN_TILES; ++tt) {
#pragma unroll
        for (int t = 0; t < 16; ++t) {
            const int idx = __shfl(bi[tt][t & 7], (t >> 3) << 4, 32);  // broadcast
            const int tok = tok0 + tt * 16 + t;
            v2f q  = *(const v2f*)(codebook + idx * EMBED_DIM + 2 * lane);
            v2f xv = *(const v2f*)(x + tok * EMBED_DIM + 2 * lane);
            // straight-through estimator computed exactly as reference: x + (q - x)
            v2f st;
            st.x = xv.x + (q.x - xv.x);
            st.y = xv.y + (q.y - xv.y);
            *(v2f*)(out_q + tok * EMBED_DIM + 2 * lane) = st;
            float dx = q.x - xv.x, dy = q.y - xv.y;
            lsum += dx * dx + dy * dy;
            if (lane == 0) out_idx[tok] = (float)idx;
        }
    }

    // Wave reduction of loss partial, then deterministic per-block partial.
#pragma unroll
    for (int off = 16; off >= 1; off >>= 1) lsum += __shfl_xor(lsum, off, 32);

    __shared__ float wsum[WAVES_PER_BLOCK];
    if (lane == 0) wsum[waveId] = lsum;
    __syncthreads();
    if (threadIdx.x == 0) {
        float s = 0.f;
#pragma unroll
        for (int w = 0; w < WAVES_PER_BLOCK; ++w) s += wsum[w];
        partials[blockIdx.x] = s;
    }
}

// ---------------------------------------------------------------------------
// Kernel 3: deterministic reduction of per-block partials -> loss scalar
// ---------------------------------------------------------------------------
__global__ void vq_loss_kernel(const float* __restrict__ partials, int nblocks,
                               float* __restrict__ loss_out, float invND) {
    __shared__ float s[256];
    float a = 0.f;
    for (int i = threadIdx.x; i < nblocks; i += 256) a += partials[i];
    s[threadIdx.x] = a;
    __syncthreads();
    for (int off = 128; off >= 1; off >>= 1) {
        if (threadIdx.x < off) s[threadIdx.x] += s[threadIdx.x + off];
        __syncthreads();
    }
    if (threadIdx.x == 0) {
        // loss = q_latent + 0.25 * e_latent = 1.25 * mean((q - x)^2)
        *loss_out = 1.25f * s[0] * invND;
    }
}

extern "C" void kernel_launch(void* const* d_in, const int* in_sizes, int n_in,
                              void* d_out, int out_size, void* d_ws, size_t ws_size,
                              hipStream_t stream) {
    const float* x        = (const float*)d_in[0];
    const float* codebook = (const float*)d_in[1];

    const int N = in_sizes[0] / EMBED_DIM;   // 262144 tokens
    const int K = in_sizes[1] / EMBED_DIM;   // 512 codes

    float* out_q    = (float*)d_out;                         // [N*64]
    float* loss_out = (float*)d_out + (size_t)N * EMBED_DIM; // [1]
    float* out_idx  = loss_out + 1;                          // [N]

    const int nblocks = N / TOKENS_PER_BLOCK;                // 2048

    float* wsf      = (float*)d_ws;
    float* partials = wsf;                                   // nblocks floats
    float* enorm    = wsf + nblocks;                         // K floats

    vq_enorm_kernel<<<(K + 255) / 256, 256, 0, stream>>>(codebook, enorm, K);
    vq_main_kernel<<<nblocks, WAVES_PER_BLOCK * 32, 0, stream>>>(
        x, codebook, enorm, out_q, out_idx, partials, K);
    vq_loss_kernel<<<1, 256, 0, stream>>>(partials, nblocks, loss_out,
                                          1.0f / (float)((size_t)N * EMBED_DIM));
}